// SparseVoxelNet_56470230008546
// MI455X (gfx1250) — compile-verified
//
#include <hip/hip_runtime.h>

// ---------------- problem constants (from reference) ----------------
#define GXD   512
#define GYD   512
#define NCH   64
#define NB    2
#define NPTS  150000
#define TOTALP (NB * NPTS)                 // 300000 points per cloud
#define SEGN  (NB * GXD * GYD)             // 524288 voxels (B*GX*GY)

typedef float v2f __attribute__((ext_vector_type(2)));
typedef float v8f __attribute__((ext_vector_type(8)));

// voxelize one point; returns validity, writes cx,cy
__device__ __forceinline__ bool voxelize(float x, float y, float z, int& cx, int& cy) {
    float fx = floorf((x - (-51.2f)) / 0.2f);
    float fy = floorf((y - (-51.2f)) / 0.2f);
    float fz = floorf((z - (-3.0f)) / 6.0f);
    cx = (int)fx; cy = (int)fy; int cz = (int)fz;
    return (cx >= 0) & (cx < GXD) & (cy >= 0) & (cy < GYD) & (cz >= 0) & (cz < 1);
}

// ---------------- zero fill ----------------
__global__ void zero_f32(float* __restrict__ p, size_t n) {
    size_t i = (size_t)blockIdx.x * blockDim.x + threadIdx.x;
    size_t stride = (size_t)gridDim.x * blockDim.x;
    for (; i < n; i += stride) p[i] = 0.0f;
}

// ---------------- pass 1: per-voxel packed stats [cnt, sx, sy, sz] ----------------
__global__ void pass1_count(const float* __restrict__ pts,
                            float* __restrict__ vox) {   // [SEGN][4]
    int p = blockIdx.x * blockDim.x + threadIdx.x;
    if (p >= TOTALP) return;
    float x = pts[3 * p + 0], y = pts[3 * p + 1], z = pts[3 * p + 2];
    int cx, cy;
    if (!voxelize(x, y, z, cx, cy)) return;
    int b = p / NPTS;
    unsigned seg = (unsigned)(b * (GXD * GYD) + cx * GYD + cy);
    // 32-bit BYTE offset (max 2^23) -> saddr + voffset atomic form
    float* vp = (float*)((char*)vox + (seg << 4));
    atomicAdd(vp + 0, 1.0f);
    atomicAdd(vp + 1, x);
    atomicAdd(vp + 2, y);
    atomicAdd(vp + 3, z);
}

// ---------------- pass 2: WMMA feature GEMM + fused affine/relu + signed scatter-mean
// One wave32 handles a tile of 16 points x 64 channels using V_WMMA_F32_16X16X4_F32.
// A (16x4 f32): lanes 0-15 hold (K=0,K=1), lanes 16-31 hold (K=2,K=3) of their point.
// B (4x16 f32): VGPR0 = row K=0 (lanes 0-15) / K=2 (lanes 16-31); VGPR1 = K=1 / K=3.
// C/D (16x16 f32, 8 VGPRs): VGPR r -> row r (lanes 0-15), row r+8 (lanes 16-31).
__global__ void pass2_wmma(const float* __restrict__ pts,
                           const float* __restrict__ vox,    // [SEGN][4] cnt,sx,sy,sz
                           const float* __restrict__ Wm,     // [8][64] row-major
                           const float* __restrict__ gamma,  // [64]
                           const float* __restrict__ beta,   // [64]
                           float* __restrict__ out,          // [SEGN*64]
                           float scale) {
    int gwave = (blockIdx.x * blockDim.x + threadIdx.x) >> 5;  // global wave id = tile id
    int lane  = threadIdx.x & 31;
    int p0 = gwave * 16;
    if (p0 >= TOTALP) return;                // uniform per wave (EXEC stays all-ones)

    int  m = lane & 15;                      // point slot this lane loads (dup in hi half)
    int  p = p0 + m;
    bool hi = lane >= 16;

    // --- per-point 8-dim features (each lane computes for point m; hi half duplicates)
    float f[8];
    int   seg = 0;
    float wsc = 0.0f;                        // scale / cnt, 0 for invalid points
    {
        float x = pts[3 * p + 0], y = pts[3 * p + 1], z = pts[3 * p + 2];
        int cx, cy;
        bool valid = voxelize(x, y, z, cx, cy);
        if (valid) {
            int b = p / NPTS;
            seg = b * (GXD * GYD) + cx * GYD + cy;
            // 32-bit byte offset -> saddr b128 gather
            float4 v = *(const float4*)((const char*)vox + ((unsigned)seg << 4));
            float ic = 1.0f / fmaxf(v.x, 1.0f);
            float mx = v.y * ic, my = v.z * ic, mz = v.w * ic;
            float ctx = (float)cx * 0.2f + 0.1f + (-51.2f);
            float cty = (float)cy * 0.2f + 0.1f + (-51.2f);
            f[0] = x;        f[1] = y;        f[2] = z;
            f[3] = x - mx;   f[4] = y - my;   f[5] = z - mz;
            f[6] = x - ctx;  f[7] = y - cty;
            wsc = scale * ic;
        } else {
            #pragma unroll
            for (int i = 0; i < 8; ++i) f[i] = 0.0f;
        }
    }

    // --- A fragments (two K=4 slabs of the 8-wide feature)
    v2f a0, a1;
    a0.x = hi ? f[2] : f[0];   a0.y = hi ? f[3] : f[1];   // K = 0..3
    a1.x = hi ? f[6] : f[4];   a1.y = hi ? f[7] : f[5];   // K = 4..7
    int krow = hi ? 2 : 0;

    // --- hoisted cross-lane redistribution: row r of D belongs to point r (+8 in hi half)
    unsigned bbase[8];                       // 32-bit BYTE offset seg*64*4 (max 2^27)
    float    ww[8];                          // scale/cnt (0 => contributes exactly 0)
    int mo0 = hi ? 8 : 0;
    #pragma unroll
    for (int r = 0; r < 8; ++r) {
        int mo   = r + mo0;
        ww[r]    = __shfl(wsc, mo, 32);
        bbase[r] = (unsigned)__shfl(seg, mo, 32) << 8;
    }

    // --- four 16-channel N-tiles
    #pragma unroll
    for (int nt = 0; nt < 4; ++nt) {
        int col = nt * 16 + m;
        v2f b0, b1;
        b0.x = Wm[(krow + 0) * NCH + col];
        b0.y = Wm[(krow + 1) * NCH + col];
        b1.x = Wm[(krow + 4) * NCH + col];
        b1.y = Wm[(krow + 5) * NCH + col];

        v8f acc = {};
        acc = __builtin_amdgcn_wmma_f32_16x16x4_f32(false, a0, false, b0,
                                                    (short)0, acc, false, false);
        acc = __builtin_amdgcn_wmma_f32_16x16x4_f32(false, a1, false, b1,
                                                    (short)0, acc, false, false);

        float    g    = gamma[col];
        float    bb   = beta[col];
        unsigned coff = (unsigned)col << 2;  // byte offset of the channel
        #pragma unroll
        for (int r = 0; r < 8; ++r) {
            // branchless: ww==0 (invalid point) adds exactly 0.0f; EXEC stays all-ones.
            // 32-bit byte offset (zext) -> saddr + voffset global_atomic_add_f32
            float h = fmaxf(acc[r] * g + bb, 0.0f);
            atomicAdd((float*)((char*)out + (bbase[r] + coff)), h * ww[r]);
        }
    }
}

// ---------------- launch ----------------
extern "C" void kernel_launch(void* const* d_in, const int* in_sizes, int n_in,
                              void* d_out, int out_size, void* d_ws, size_t ws_size,
                              hipStream_t stream) {
    const float* pch1  = (const float*)d_in[0];   // history  -> scale -0.5
    const float* pc0   = (const float*)d_in[1];   // t=1      -> scale -0.5
    const float* pc1   = (const float*)d_in[2];   // reference-> scale +1.0
    const float* Wm    = (const float*)d_in[3];
    const float* gamma = (const float*)d_in[4];
    const float* beta  = (const float*)d_in[5];
    float* out = (float*)d_out;
    float* wsf = (float*)d_ws;

    // ws layout: 3 clouds x SEGN x 4 floats (cnt,sx,sy,sz) = 25 MB (fits L2 easily)
    const float* clouds[3] = {pc1, pch1, pc0};
    const float  scales[3] = {1.0f, -0.5f, -0.5f};

    size_t n_out = (size_t)SEGN * NCH;       // 33.55M floats (128 MB)
    size_t n_ws  = (size_t)3 * SEGN * 4;

    zero_f32<<<2048, 256, 0, stream>>>(out, n_out);
    zero_f32<<<1024, 256, 0, stream>>>(wsf, n_ws);

    int p1_blocks = (TOTALP + 255) / 256;
    for (int c = 0; c < 3; ++c) {
        pass1_count<<<p1_blocks, 256, 0, stream>>>(
            clouds[c], wsf + (size_t)c * SEGN * 4);
    }

    int tiles     = (TOTALP + 15) / 16;      // 18750 wave-tiles
    int p2_blocks = (tiles + 7) / 8;         // 8 waves per 256-thread block
    for (int c = 0; c < 3; ++c) {
        pass2_wmma<<<p2_blocks, 256, 0, stream>>>(
            clouds[c], wsf + (size_t)c * SEGN * 4,
            Wm, gamma, beta, out, scales[c]);
    }
}